// ControlledChaoticOscillator_71683004170563
// MI455X (gfx1250) — compile-verified
//
#include <hip/hip_runtime.h>

typedef float v2f __attribute__((ext_vector_type(2)));
typedef float v8f __attribute__((ext_vector_type(8)));

constexpr int B_ = 64, S_ = 1024, D_ = 512, H_ = 1024;
constexpr int BS_ = B_ * S_;          // 65536 rows
constexpr float HSTEP = 0.001f;       // INTEGRATION_TIME / INTEGRATION_STEPS

// ---------------------------------------------------------------------------
// Kernel 0: fold the 3x3 coupling into the input projection:
//   W2 = C @ W_in  (3 x 512),  b2 = C @ b_in  (3)
// so that uc = control @ C^T = x @ W2^T + b2 can come straight out of phase 1.
// ---------------------------------------------------------------------------
__global__ void prep_kernel(const float* __restrict__ W_in,
                            const float* __restrict__ b_in,
                            const float* __restrict__ C,
                            float* __restrict__ W2, float* __restrict__ b2) {
  int d = threadIdx.x;            // 512 threads
#pragma unroll
  for (int i = 0; i < 3; ++i) {
    W2[i * D_ + d] = C[i * 3 + 0] * W_in[0 * D_ + d]
                   + C[i * 3 + 1] * W_in[1 * D_ + d]
                   + C[i * 3 + 2] * W_in[2 * D_ + d];
  }
  if (d < 3) {
    b2[d] = C[d * 3 + 0] * b_in[0] + C[d * 3 + 1] * b_in[1] + C[d * 3 + 2] * b_in[2];
  }
}

// ---------------------------------------------------------------------------
// Kernel 1: fp32 WMMA GEMM, M=65536 (b,s rows), N=16 (cols 0-2 = control,
// cols 4-6 = uc = control @ C^T), K=512.  One wave per 16-row tile.
// HBM-bound (134 MB read of x).
// ---------------------------------------------------------------------------
__global__ void __launch_bounds__(128)
gemm_in_kernel(const float* __restrict__ x,
               const float* __restrict__ W_in, const float* __restrict__ b_in,
               const float* __restrict__ W2,   const float* __restrict__ b2,
               float* __restrict__ uc, float* __restrict__ control0) {
  __shared__ float Bl[D_ * 16];   // combined B matrix, [k][n] layout, 32 KB

  const int tid = threadIdx.x;
  for (int idx = tid; idx < D_ * 16; idx += 128) {
    int k = idx >> 4, n = idx & 15;
    float v = 0.0f;
    if (n < 3)              v = W_in[n * D_ + k];
    else if (n >= 4 && n < 7) v = W2[(n - 4) * D_ + k];
    Bl[idx] = v;
  }
  __syncthreads();

  const int wave  = tid >> 5;
  const int lane  = tid & 31;
  const int tile  = blockIdx.x * 4 + wave;       // 4096 tiles total
  const int r0    = tile * 16;
  const int n     = lane & 15;                   // A row within tile / B column
  const int khalf = lane >> 4;                   // 0: K=0,1   1: K=2,3
  const float* xrow = x + (size_t)(r0 + n) * D_ + 2 * khalf;

  v8f acc = {};
  for (int kb = 0; kb < D_; kb += 4) {
    if ((kb & 63) == 0) __builtin_prefetch(xrow + kb + 64, 0, 0);  // global_prefetch_b8
    v2f a;                                   // A 16x4 fragment (ISA layout)
    a.x = xrow[kb];
    a.y = xrow[kb + 1];
    v2f b;                                   // B 4x16 fragment from LDS (conflict-free)
    int kk = kb + 2 * khalf;
    b.x = Bl[kk * 16 + n];
    b.y = Bl[(kk + 1) * 16 + n];
    acc = __builtin_amdgcn_wmma_f32_16x16x4_f32(false, a, false, b,
                                                (short)0, acc, false, false);
  }

  // Epilogue: bias + scatter. D layout: lane -> col n, VGPR v -> row v + 8*khalf.
  float bias = 0.0f;
  if (n < 3)                bias = b_in[n];
  else if (n >= 4 && n < 7) bias = b2[n - 4];
#pragma unroll
  for (int v = 0; v < 8; ++v) {
    int r = r0 + v + 8 * khalf;
    float val = acc[v] + bias;
    if (n >= 4 && n < 7)                    uc[(size_t)r * 3 + (n - 4)] = val;
    if (n < 3 && ((r & (S_ - 1)) == 0))     control0[(r >> 10) * 3 + n] = val;  // state0 = control[:,0]
  }
}

// ---------------------------------------------------------------------------
// Kernel 2: the sequential RK4 Lorenz scan. Latency-bound, parallelism = 64
// trajectories -> 64 threads (2 wave32), state in registers, RK4 unrolled.
// ---------------------------------------------------------------------------
__global__ void scan_kernel(const float* __restrict__ uc,
                            const float* __restrict__ control0,
                            const float* __restrict__ sigma_p,
                            const float* __restrict__ rho_p,
                            const float* __restrict__ beta_p,
                            float* __restrict__ states) {
  const int b = threadIdx.x;                 // 64 threads
  const float sigma = sigma_p[0], rho = rho_p[0], beta = beta_p[0];
  const float h = HSTEP, hh = 0.5f * HSTEP, h6 = HSTEP * (1.0f / 6.0f);

  float sx = control0[b * 3 + 0];
  float sy = control0[b * 3 + 1];
  float sz = control0[b * 3 + 2];
  const float* ucb = uc + (size_t)b * S_ * 3;
  float*       stb = states + (size_t)b * S_ * 3;

  for (int s = 0; s < S_; ++s) {
    const float u0 = ucb[s * 3 + 0];
    const float u1 = ucb[s * 3 + 1];
    const float u2 = ucb[s * 3 + 2];
#pragma unroll
    for (int it = 0; it < 10; ++it) {
      float k1x = __builtin_fmaf(sigma, sy - sx, u0);
      float k1y = __builtin_fmaf(sx, rho - sz, u1 - sy);
      float k1z = __builtin_fmaf(sx, sy, __builtin_fmaf(-beta, sz, u2));
      float ax = __builtin_fmaf(hh, k1x, sx), ay = __builtin_fmaf(hh, k1y, sy), az = __builtin_fmaf(hh, k1z, sz);
      float k2x = __builtin_fmaf(sigma, ay - ax, u0);
      float k2y = __builtin_fmaf(ax, rho - az, u1 - ay);
      float k2z = __builtin_fmaf(ax, ay, __builtin_fmaf(-beta, az, u2));
      float bx = __builtin_fmaf(hh, k2x, sx), by = __builtin_fmaf(hh, k2y, sy), bz = __builtin_fmaf(hh, k2z, sz);
      float k3x = __builtin_fmaf(sigma, by - bx, u0);
      float k3y = __builtin_fmaf(bx, rho - bz, u1 - by);
      float k3z = __builtin_fmaf(bx, by, __builtin_fmaf(-beta, bz, u2));
      float cx = __builtin_fmaf(h, k3x, sx), cy = __builtin_fmaf(h, k3y, sy), cz = __builtin_fmaf(h, k3z, sz);
      float k4x = __builtin_fmaf(sigma, cy - cx, u0);
      float k4y = __builtin_fmaf(cx, rho - cz, u1 - cy);
      float k4z = __builtin_fmaf(cx, cy, __builtin_fmaf(-beta, cz, u2));
      sx = __builtin_fmaf(h6, k1x + 2.0f * k2x + 2.0f * k3x + k4x, sx);
      sy = __builtin_fmaf(h6, k1y + 2.0f * k2y + 2.0f * k3y + k4y, sy);
      sz = __builtin_fmaf(h6, k1z + 2.0f * k2z + 2.0f * k3z + k4z, sz);
    }
    stb[s * 3 + 0] = sx;
    stb[s * 3 + 1] = sy;
    stb[s * 3 + 2] = sz;
  }
}

// ---------------------------------------------------------------------------
// Kernel 3: fp32 WMMA GEMM, M=65536, N=1024, K=3 (padded to 4).
// Exactly one v_wmma_f32_16x16x4_f32 per 16x16 output tile; store-bound (268 MB).
// ---------------------------------------------------------------------------
__global__ void __launch_bounds__(128)
gemm_out_kernel(const float* __restrict__ states,
                const float* __restrict__ W_out,
                const float* __restrict__ b_out,
                float* __restrict__ out) {
  const int tid    = threadIdx.x;
  const int wave   = tid >> 5;
  const int lane   = tid & 31;
  const int tileId = blockIdx.x * 4 + wave;      // 262144 tiles total
  const int rowT   = tileId >> 6;                // 64 column tiles per row
  const int colT   = tileId & 63;
  const int r0 = rowT * 16, h0 = colT * 16;
  const int n = lane & 15, khalf = lane >> 4;

  v2f a, bfrag;
  const size_t arow = (size_t)(r0 + n) * 3;
  a.x = states[arow + 2 * khalf];                     // K = 0 or 2
  a.y = khalf ? 0.0f : states[arow + 1];              // K = 1 (K=3 padded with 0)
  const float* wrow = W_out + (size_t)(h0 + n) * 3;
  bfrag.x = wrow[2 * khalf];
  bfrag.y = khalf ? 0.0f : wrow[1];

  v8f c = {};
  c = __builtin_amdgcn_wmma_f32_16x16x4_f32(false, a, false, bfrag,
                                            (short)0, c, false, false);

  const float bias = b_out[h0 + n];
#pragma unroll
  for (int v = 0; v < 8; ++v) {
    int r = r0 + v + 8 * khalf;
    out[(size_t)r * H_ + h0 + n] = c[v] + bias;
  }
}

// ---------------------------------------------------------------------------
extern "C" void kernel_launch(void* const* d_in, const int* in_sizes, int n_in,
                              void* d_out, int out_size, void* d_ws, size_t ws_size,
                              hipStream_t stream) {
  const float* x     = (const float*)d_in[0];
  const float* W_in  = (const float*)d_in[1];
  const float* b_in  = (const float*)d_in[2];
  const float* C     = (const float*)d_in[3];
  const float* W_out = (const float*)d_in[4];
  const float* b_out = (const float*)d_in[5];
  const float* sigma = (const float*)d_in[6];
  const float* rho   = (const float*)d_in[7];
  const float* beta  = (const float*)d_in[8];
  float* out = (float*)d_out;

  float* ws       = (float*)d_ws;
  float* uc       = ws;                       // 65536*3 floats
  float* states   = ws + (size_t)BS_ * 3;     // 65536*3 floats
  float* W2       = ws + (size_t)BS_ * 6;     // 3*512
  float* b2       = W2 + 3 * D_;              // 3 (+1 pad)
  float* control0 = b2 + 4;                   // 64*3

  prep_kernel<<<1, D_, 0, stream>>>(W_in, b_in, C, W2, b2);
  gemm_in_kernel<<<BS_ / 16 / 4, 128, 0, stream>>>(x, W_in, b_in, W2, b2, uc, control0);
  scan_kernel<<<1, B_, 0, stream>>>(uc, control0, sigma, rho, beta, states);
  gemm_out_kernel<<<(BS_ / 16) * (H_ / 16) / 4, 128, 0, stream>>>(states, W_out, b_out, out);
}